// SparseDeformableMambaBlock_39633958208136
// MI455X (gfx1250) — compile-verified
//
#include <hip/hip_runtime.h>
#include <cstdint>
#include <cstddef>

// Sizes fixed by the reference
#define BATCH   8
#define LSEQ    4096
#define DIMC    256            // DIM
#define DMODEL  512            // D = DIM*EXPAND
#define NSTATE  16
#define KTOP    1228           // int(4096*0.3)
#define M1      (BATCH * LSEQ) // 32768 rows for GEMM1
#define M2      (BATCH * KTOP) // 9824 rows for GEMM2 (== 614*16 exactly)
#define BSTRIDE 36             // LDS dword stride for B tiles (conflict-free)

typedef __attribute__((ext_vector_type(2))) float v2f;
typedef __attribute__((ext_vector_type(8))) float v8f;

// D = A(16x4,f32) * B(4x16,f32) + C(16x16,f32)   (wave32, exact f32)
static __device__ __forceinline__ v8f wmma4(v2f a, v2f b, v8f c) {
  return __builtin_amdgcn_wmma_f32_16x16x4_f32(false, a, false, b, (short)0, c,
                                               false, false);
}

// Async-stage a 128col x 32k B-chunk of W (stored N x K) into LDS via the
// CDNA5 async copy path (ASYNCcnt). Each of the 256 threads moves 16 floats
// as 4x global_load_async_to_lds_b128. LDS layout: [col][k], stride BSTRIDE.
static __device__ __forceinline__ void stage_b_async(
    const float* __restrict__ W, int n0, int kc, int kdim, float* dst) {
  const int t = threadIdx.x;
  const int col = t & 127;
  const int khalf = (t >> 7) << 4;  // 0 or 16
  const float* g = W + (size_t)(n0 + col) * kdim + kc + khalf;
  // low 32 bits of a flat LDS pointer == LDS byte offset (aperture low bits)
  unsigned lds0 = (unsigned)(unsigned long long)(const void*)(dst + col * BSTRIDE + khalf);
#pragma unroll
  for (int i = 0; i < 4; ++i) {
    unsigned long long ga = (unsigned long long)(const void*)(g + i * 4);
    unsigned la = lds0 + i * 16;
    asm volatile("global_load_async_to_lds_b128 %0, %1, off"
                 :: "v"(la), "v"(ga) : "memory");
  }
}

// ---------------------------------------------------------------- DyT norm
__global__ __launch_bounds__(256) void k_dyt(const float* __restrict__ x,
                                             const float* __restrict__ alpha,
                                             const float* __restrict__ w,
                                             const float* __restrict__ bb,
                                             float* __restrict__ xn) {
  int i = blockIdx.x * 256 + threadIdx.x;         // float4 index
  float a = alpha[0];
  float4 v = ((const float4*)x)[i];
  int c = (i * 4) & (DIMC - 1);                   // DIMC % 4 == 0
  float4 r;
  r.x = tanhf(a * v.x) * w[c + 0] + bb[c + 0];
  r.y = tanhf(a * v.y) * w[c + 1] + bb[c + 1];
  r.z = tanhf(a * v.z) * w[c + 2] + bb[c + 2];
  r.w = tanhf(a * v.w) * w[c + 3] + bb[c + 3];
  ((float4*)xn)[i] = r;
}

// ---------------------------------------------------------------- GEMM1
// xp[m][n] = sum_k xn[m][k] * Win[n][k] + bin[n]   (M1 x 256 -> 512)
// Block: 8 waves, 128 rows x 128 cols. B staged in LDS (double-buffered,
// async copies). Wave: 16 rows x 8 wmma tiles, A as b64 pairs from global.
__global__ __launch_bounds__(256) void k_gemm1(const float* __restrict__ xn,
                                               const float* __restrict__ Win,
                                               const float* __restrict__ bin,
                                               float* __restrict__ xp) {
  __shared__ float bbuf[2][128 * BSTRIDE];  // 2 x 18 KB
  const int lane = threadIdx.x & 31;
  const int wv   = threadIdx.x >> 5;
  const int lr   = lane & 15;
  const int khi  = (lane >> 4) << 1;  // 0 or 2
  const int hi8  = (lane >> 4) << 3;  // 0 or 8
  const int m0 = blockIdx.y * 128 + wv * 16;
  const int n0 = blockIdx.x * 128;

  v8f acc[8];
#pragma unroll
  for (int nt = 0; nt < 8; ++nt) {
    float bv = bin[n0 + nt * 16 + lr];
#pragma unroll
    for (int r = 0; r < 8; ++r) acc[nt][r] = bv;
  }

  stage_b_async(Win, n0, 0, DIMC, bbuf[0]);
  const float* arow = xn + (size_t)(m0 + lr) * DIMC + khi;
  int buf = 0;
  for (int kc = 0; kc < DIMC; kc += 32) {
    if (kc + 32 < DIMC) {
      stage_b_async(Win, n0, kc + 32, DIMC, bbuf[buf ^ 1]);
      asm volatile("s_wait_asynccnt 0x4" ::: "memory");  // chunk kc landed
    } else {
      asm volatile("s_wait_asynccnt 0x0" ::: "memory");  // last chunk landed
    }
    __syncthreads();
    const float* bb = bbuf[buf];
#pragma unroll 2
    for (int k = 0; k < 32; k += 4) {
      v2f a = *(const v2f*)(arow + kc + k);
#pragma unroll
      for (int nt = 0; nt < 8; ++nt) {
        v2f b = *(const v2f*)(bb + (nt * 16 + lr) * BSTRIDE + k + khi);
        acc[nt] = wmma4(a, b, acc[nt]);
      }
    }
    __syncthreads();  // before overwriting this buffer
    buf ^= 1;
  }
#pragma unroll
  for (int nt = 0; nt < 8; ++nt)
#pragma unroll
    for (int r = 0; r < 8; ++r)
      xp[(size_t)(m0 + hi8 + r) * DMODEL + n0 + nt * 16 + lr] = acc[nt][r];
}

// ---------------------------------------------------------------- similarity
// sim ordering-equivalent: dot(x_proj,center) * rsqrt(||x||^2+eps)
// (dropping the positive per-batch 1/||center|| keeps top-k ordering)
__global__ __launch_bounds__(256) void k_sim(const float* __restrict__ xp,
                                             float* __restrict__ sim) {
  __shared__ float cen[DMODEL];
  const int b = (blockIdx.x * 8) >> 12;  // 8 rows per block, 4096 rows/batch
  const float* crow = xp + ((size_t)b * LSEQ + LSEQ / 2) * DMODEL;
  for (int e = threadIdx.x; e < DMODEL; e += 256) cen[e] = crow[e];
  __syncthreads();
  const int wv = threadIdx.x >> 5, lane = threadIdx.x & 31;
  const int row = blockIdx.x * 8 + wv;
  const float* xr = xp + (size_t)row * DMODEL;
  float xx = 0.f, xc = 0.f;
  for (int e = lane; e < DMODEL; e += 32) {
    float v = xr[e];
    xx += v * v;
    xc += v * cen[e];
  }
#pragma unroll
  for (int off = 16; off > 0; off >>= 1) {
    xx += __shfl_xor(xx, off, 32);
    xc += __shfl_xor(xc, off, 32);
  }
  if (lane == 0) sim[row] = xc * rsqrtf(xx + 1e-24f);
}

// ---------------------------------------------------------------- top-k
// Bitonic sort of 4096 packed keys per batch; reproduces lax.top_k order:
// descending by value, stable (lower index first on ties).
__global__ __launch_bounds__(1024) void k_topk(const float* __restrict__ sim,
                                               int* __restrict__ topk) {
  __shared__ unsigned long long key[LSEQ];  // 32 KB
  const int b = blockIdx.x, tid = threadIdx.x;
  for (int i = tid; i < LSEQ; i += 1024) {
    unsigned u = __float_as_uint(sim[(size_t)b * LSEQ + i]);
    u = (u & 0x80000000u) ? ~u : (u | 0x80000000u);  // order-preserving map
    key[i] = ((unsigned long long)u << 32) | (unsigned)(LSEQ - 1 - i);
  }
  __syncthreads();
  for (unsigned kk = 2; kk <= LSEQ; kk <<= 1) {
    for (unsigned j = kk >> 1; j > 0; j >>= 1) {
      for (unsigned i = tid; i < LSEQ; i += 1024) {
        unsigned ixj = i ^ j;
        if (ixj > i) {
          unsigned long long a = key[i], c = key[ixj];
          bool sw = ((i & kk) == 0) ? (a < c) : (a > c);  // overall descending
          if (sw) { key[i] = c; key[ixj] = a; }
        }
      }
      __syncthreads();
    }
  }
  for (int i = tid; i < KTOP; i += 1024)
    topk[(size_t)b * KTOP + i] = (LSEQ - 1) - (int)(key[i] & 0xFFFFFFFFu);
}

// ------------------------------------------------- gather + depthwise conv
__global__ __launch_bounds__(256) void k_conv(const float* __restrict__ xp,
                                              const int* __restrict__ topk,
                                              const float* __restrict__ cw,
                                              float* __restrict__ xc) {
  const int d = blockIdx.x * 256 + threadIdx.x;
  const int t = blockIdx.y, b = blockIdx.z;
  float acc = 0.f;
#pragma unroll
  for (int c = 0; c < 4; ++c) {
    int tau = t + c - 3;  // left pad D_CONV-1 = 3
    if (tau >= 0) {
      int l = topk[b * KTOP + tau];
      acc += cw[d * 4 + c] * xp[((size_t)b * LSEQ + l) * DMODEL + d];
    }
  }
  xc[((size_t)b * KTOP + t) * DMODEL + d] = acc;
}

// ---------------------------------------------------------------- scan
// Per wave: 16 (b,d) rows, state H(16x16) held in the WMMA accumulator.
// Step: H = H * A^T + outer(x_t, sigB)  as 4 chained F32 16x16x4 WMMAs.
// D-layout -> A-operand-layout recirculation via a wave-private LDS tile
// (row stride 20 dwords => conflict-free), ordered with s_wait_dscnt.
__global__ __launch_bounds__(256) void k_scan(const float* __restrict__ xc,
                                              const float* __restrict__ A,
                                              const float* __restrict__ Bp,
                                              const float* __restrict__ Cp,
                                              float* __restrict__ outs) {
  __shared__ float tile[8 * 16 * 20];  // 10 KB, one 16x20 tile per wave
  const int lane = threadIdx.x & 31;
  const int wv   = threadIdx.x >> 5;
  const int gw   = blockIdx.x * 8 + wv;  // 0..255 global wave
  const int b    = gw >> 5;              // 32 waves per batch
  const int d0   = (gw & 31) * 16;
  const int n    = lane & 15;
  const int k0   = (lane >> 4) * 2;
  const int hi8  = (lane >> 4) * 8;
  float* lds = tile + wv * (16 * 20);

  // B operand = A^T, constant: Bop[k=j][n=i] = A[i][j] -> contiguous b64 pairs
  v2f Bq[4];
#pragma unroll
  for (int q = 0; q < 4; ++q) Bq[q] = *(const v2f*)(A + n * 16 + q * 4 + k0);

  const float sB = 1.f / (1.f + expf(-Bp[n]));  // sigB[col]
  float sC[8];                                  // sigC[d0+n][hi8 .. hi8+7]
#pragma unroll
  for (int i = 0; i < 8; ++i)
    sC[i] = 1.f / (1.f + expf(-Cp[(d0 + n) * NSTATE + hi8 + i]));

  v2f Aq[4];  // H in A-operand layout; starts at zero
#pragma unroll
  for (int q = 0; q < 4; ++q) { Aq[q][0] = 0.f; Aq[q][1] = 0.f; }

  const float* xrow = xc + (size_t)b * KTOP * DMODEL + d0;
  float* orow = outs + (size_t)b * KTOP * DMODEL + d0;

  for (int t = 0; t < KTOP; ++t) {
    __builtin_prefetch(xrow + DMODEL, 0, 1);  // global_prefetch_b8: next x_t
    // C = outer(x_t, sigB) in accumulator layout; x_t as two b128 loads
    float4 x0 = *(const float4*)(xrow + hi8);
    float4 x1 = *(const float4*)(xrow + hi8 + 4);
    v8f h;
    h[0] = x0.x * sB; h[1] = x0.y * sB; h[2] = x0.z * sB; h[3] = x0.w * sB;
    h[4] = x1.x * sB; h[5] = x1.y * sB; h[6] = x1.z * sB; h[7] = x1.w * sB;
    // H_new = H * A^T + C  (K=16 as 4 chained K=4 WMMAs)
    h = wmma4(Aq[0], Bq[0], h);
    h = wmma4(Aq[1], Bq[1], h);
    h = wmma4(Aq[2], Bq[2], h);
    h = wmma4(Aq[3], Bq[3], h);
    // D layout -> LDS (wave-private; DS ops are in-order per wave)
#pragma unroll
    for (int r = 0; r < 8; ++r) lds[(hi8 + r) * 20 + n] = h[r];
    asm volatile("s_wait_dscnt 0x0" ::: "memory");
    // reload as next step's A operand
#pragma unroll
    for (int q = 0; q < 4; ++q)
      Aq[q] = *(const v2f*)(lds + n * 20 + q * 4 + k0);
    // out[t][d0+n] = sum_i H[n][i]*sigC ; halves combine via shfl_xor(16)
    float part = 0.f;
#pragma unroll
    for (int i = 0; i < 8; ++i) part += lds[n * 20 + hi8 + i] * sC[i];
    part += __shfl_xor(part, 16, 32);
    if (lane < 16) orow[n] = part;
    xrow += DMODEL;
    orow += DMODEL;
  }
}

// ---------------------------------------------------------------- residual
__global__ __launch_bounds__(256) void k_copy(const float* __restrict__ x,
                                              float* __restrict__ out) {
  size_t i = (size_t)blockIdx.x * 256 + threadIdx.x;
  ((float4*)out)[i] = ((const float4*)x)[i];
}

// ------------------------------------------- GEMM2 + scatter + residual
// out[b, topk[b][t], c] = x[...] + sum_d scanout[b,t,d]*Wout[c][d] + bout[c]
// Same LDS-staged structure as GEMM1; tail waves are clamped (no divergent
// return around __syncthreads), stores are guarded instead.
__global__ __launch_bounds__(256) void k_gemm2(const float* __restrict__ so,
                                               const float* __restrict__ Wout,
                                               const float* __restrict__ bout,
                                               const int* __restrict__ topk,
                                               const float* __restrict__ x,
                                               float* __restrict__ out) {
  __shared__ float bbuf[2][128 * BSTRIDE];
  const int lane = threadIdx.x & 31;
  const int wv   = threadIdx.x >> 5;
  const int lr   = lane & 15;
  const int khi  = (lane >> 4) << 1;
  const int hi8  = (lane >> 4) << 3;
  const int mt_raw = blockIdx.y * 8 + wv;       // wave-uniform
  const int valid  = mt_raw < (M2 / 16);
  const int mt = valid ? mt_raw : (M2 / 16 - 1);
  const int n0 = blockIdx.x * 128;
  const int m0 = mt * 16;

  v8f acc[8];
#pragma unroll
  for (int nt = 0; nt < 8; ++nt) {
    float bv = bout[n0 + nt * 16 + lr];
#pragma unroll
    for (int r = 0; r < 8; ++r) acc[nt][r] = bv;
  }

  stage_b_async(Wout, n0, 0, DMODEL, bbuf[0]);
  const float* arow = so + (size_t)(m0 + lr) * DMODEL + khi;
  int buf = 0;
  for (int kc = 0; kc < DMODEL; kc += 32) {
    if (kc + 32 < DMODEL) {
      stage_b_async(Wout, n0, kc + 32, DMODEL, bbuf[buf ^ 1]);
      asm volatile("s_wait_asynccnt 0x4" ::: "memory");
    } else {
      asm volatile("s_wait_asynccnt 0x0" ::: "memory");
    }
    __syncthreads();
    const float* bb = bbuf[buf];
#pragma unroll 2
    for (int k = 0; k < 32; k += 4) {
      v2f a = *(const v2f*)(arow + kc + k);
#pragma unroll
      for (int nt = 0; nt < 8; ++nt) {
        v2f b = *(const v2f*)(bb + (nt * 16 + lr) * BSTRIDE + k + khi);
        acc[nt] = wmma4(a, b, acc[nt]);
      }
    }
    __syncthreads();
    buf ^= 1;
  }
  if (valid) {
#pragma unroll
    for (int r = 0; r < 8; ++r) {
      int mg = m0 + hi8 + r;
      int bb = mg / KTOP;
      int tt = mg - bb * KTOP;
      int l = topk[bb * KTOP + tt];
      size_t base = ((size_t)bb * LSEQ + l) * DIMC;
#pragma unroll
      for (int nt = 0; nt < 8; ++nt) {
        size_t idx = base + n0 + nt * 16 + lr;
        out[idx] = x[idx] + acc[nt][r];
      }
    }
  }
}

// ---------------------------------------------------------------- launch
extern "C" void kernel_launch(void* const* d_in, const int* in_sizes, int n_in,
                              void* d_out, int out_size, void* d_ws,
                              size_t ws_size, hipStream_t stream) {
  const float* x     = (const float*)d_in[0];
  const float* alpha = (const float*)d_in[1];
  const float* dyt_w = (const float*)d_in[2];
  const float* dyt_b = (const float*)d_in[3];
  const float* Win   = (const float*)d_in[4];
  const float* bin   = (const float*)d_in[5];
  const float* Wout  = (const float*)d_in[6];
  const float* bout  = (const float*)d_in[7];
  const float* A     = (const float*)d_in[8];
  const float* Bp    = (const float*)d_in[9];
  const float* Cp    = (const float*)d_in[10];
  const float* cw    = (const float*)d_in[11];
  float* out = (float*)d_out;

  // workspace layout (~142 MB total)
  char* ws = (char*)d_ws;
  size_t off = 0;
  auto alloc = [&](size_t bytes) -> void* {
    void* p = ws + off;
    off = (off + bytes + 255) & ~(size_t)255;
    return p;
  };
  float* xn   = (float*)alloc((size_t)M1 * DIMC * sizeof(float));   // 33.6 MB
  float* xp   = (float*)alloc((size_t)M1 * DMODEL * sizeof(float)); // 67.1 MB
  float* sim  = (float*)alloc((size_t)M1 * sizeof(float));
  int*   topk = (int*)alloc((size_t)BATCH * KTOP * sizeof(int));
  float* xc   = (float*)alloc((size_t)M2 * DMODEL * sizeof(float)); // 20.1 MB
  float* so   = (float*)alloc((size_t)M2 * DMODEL * sizeof(float)); // 20.1 MB
  (void)ws_size; (void)in_sizes; (void)n_in; (void)out_size;

  k_dyt<<<(M1 * DIMC) / 1024, 256, 0, stream>>>(x, alpha, dyt_w, dyt_b, xn);
  k_gemm1<<<dim3(DMODEL / 128, M1 / 128), 256, 0, stream>>>(xn, Win, bin, xp);
  k_sim<<<M1 / 8, 256, 0, stream>>>(xp, sim);
  k_topk<<<BATCH, 1024, 0, stream>>>(sim, topk);
  k_conv<<<dim3(DMODEL / 256, KTOP, BATCH), 256, 0, stream>>>(xp, topk, cw, xc);
  k_copy<<<(M1 * DIMC) / 1024, 256, 0, stream>>>(x, out);
  k_scan<<<32, 256, 0, stream>>>(xc, A, Bp, Cp, so);
  k_gemm2<<<dim3(DIMC / 128, (M2 / 16 + 7) / 8), 256, 0, stream>>>(so, Wout, bout,
                                                                   topk, x, out);
}